// Qwen2RWKV6cSimple_68075231642248
// MI455X (gfx1250) — compile-verified
//
#include <hip/hip_runtime.h>

// Chunked simple-GLA forward for gfx1250 (MI455X).
// fp32 math via V_WMMA_F32_16X16X4_F32; async global->LDS double buffering
// (GLOBAL_LOAD_ASYNC_TO_LDS_B128 + s_wait_asynccnt); wave shfl-scan for gates.
//
// Grid: one workgroup per (b*h, v-slice of 16). 128 threads = 4 wave32,
// wave w owns output row-tile m=w and state k-tiles {2w, 2w+1}.
// Per chunk: o = e^{gc}*scale * q@S  +  tril(q@k^T * e^{gc_i-gc_j})*scale @ v
//            S = e^{glast} * S + k^T @ (v * e^{glast-gc}).

typedef float v2f __attribute__((ext_vector_type(2)));
typedef float v8f __attribute__((ext_vector_type(8)));

#define T_LEN 4096
#define KD    128
#define VD    128
#define BT    64
#define NT    (T_LEN / BT)
#define VS    16          // v-slice per workgroup
#define NVB   (VD / VS)   // 8 v-slices

// LDS row strides (floats): padded so 16 consecutive rows hit distinct banks
#define QS  132   // q/k rows (128 + 4)
#define VST 20    // v/state rows (16 + 4)
#define SSL 20    // s-slab rows (16 + 4)

#define NASYNC 34  // async b128 ops issued per thread per chunk (16 q + 16 k + 2 v)

__device__ __forceinline__ v8f wmma4(v2f a, v2f b, v8f c) {
  // D(16x16,f32) = A(16x4,f32) * B(4x16,f32) + C
  return __builtin_amdgcn_wmma_f32_16x16x4_f32(
      false, a, false, b, (short)0, c, false, false);
}

// A fragment (16x4): lanes 0-15 -> M=lane, VGPR0/1 = K=0/1 ; lanes 16-31 -> K=2/3.
__device__ __forceinline__ v2f ld_a(const float* m, int stride, int row0, int k0,
                                    int lr, int hf) {
  const float* p = m + (row0 + lr) * stride + k0 + 2 * hf;
  v2f a; a.x = p[0]; a.y = p[1];
  return a;
}

// B fragment (4x16): VGPR r, lanes 0-15 -> K=r ; lanes 16-31 -> K=r+2.
__device__ __forceinline__ v2f ld_b(const float* m, int stride, int k0, int n0,
                                    int lr, int hf) {
  v2f b;
  b.x = m[(k0 + 2 * hf) * stride + n0 + lr];
  b.y = m[(k0 + 2 * hf + 1) * stride + n0 + lr];
  return b;
}

// Async global -> LDS, 16B per lane. LDS byte address = low 32 bits of generic ptr.
__device__ __forceinline__ void async_b128(unsigned lds_off, const void* gptr) {
  asm volatile("global_load_async_to_lds_b128 %0, %1, off"
               :: "v"(lds_off), "v"(gptr) : "memory");
}

__device__ __forceinline__ void issue_chunk_loads(
    const float* qb, const float* kb, const float* vp, int t0,
    float* Sq, float* Sk, float* Sv, int tid) {
#pragma unroll
  for (int it = 0; it < 16; ++it) {
    int f4 = tid + 128 * it;           // 2048 b128 blocks over q (and k)
    int row = f4 >> 5;
    int c4 = (f4 & 31) << 2;
    async_b128((unsigned)(uintptr_t)&Sq[row * QS + c4],
               qb + (size_t)(t0 + row) * KD + c4);
    async_b128((unsigned)(uintptr_t)&Sk[row * QS + c4],
               kb + (size_t)(t0 + row) * KD + c4);
  }
#pragma unroll
  for (int it = 0; it < 2; ++it) {
    int f4 = tid + 128 * it;           // 256 b128 blocks over v slice
    int row = f4 >> 2;
    int c4 = (f4 & 3) << 2;
    async_b128((unsigned)(uintptr_t)&Sv[row * VST + c4],
               vp + (size_t)(t0 + row) * VD + c4);
  }
}

__global__ __launch_bounds__(128, 1)
void gla_fwd_kernel(const float* __restrict__ q, const float* __restrict__ k,
                    const float* __restrict__ v, const float* __restrict__ g,
                    float* __restrict__ out) {
  __shared__ __align__(16) float SqB[2 * BT * QS];
  __shared__ __align__(16) float SkB[2 * BT * QS];
  __shared__ __align__(16) float SvB[2 * BT * VST];
  __shared__ __align__(16) float St[KD * VST];      // running state [K][VS]
  __shared__ __align__(16) float Ss[4 * 16 * SSL];  // per-wave masked-s slab
  __shared__ float EgS[BT];  // exp(gc_i) * scale
  __shared__ float Rg[BT];   // exp(-gc_j)
  __shared__ float Wv[BT];   // exp(glast - gc_j)
  __shared__ float Dec;      // exp(glast)

  const int tid  = threadIdx.x;
  const int wave = tid >> 5;
  const int lane = tid & 31;
  const int lr   = lane & 15;
  const int hf   = lane >> 4;

  const int bh = blockIdx.x / NVB;
  const int vb = blockIdx.x % NVB;
  const int v0 = vb * VS;

  const int m0 = wave * 16;            // wave = M tile
  const float scale = 0.08838834764831845f;  // 128^-0.5

  const float* qb = q + (size_t)bh * T_LEN * KD;
  const float* kb = k + (size_t)bh * T_LEN * KD;
  const float* vp = v + (size_t)bh * T_LEN * VD + v0;
  const float* gb = g + (size_t)bh * T_LEN;
  float*       ob = out + (size_t)bh * T_LEN * VD + v0;

  float* slab = Ss + wave * 16 * SSL;

  // zero state; prefetch chunk 0 (buffer 0)
  for (int i = tid; i < KD * VST; i += 128) St[i] = 0.0f;
  issue_chunk_loads(qb, kb, vp, 0, SqB, SkB, SvB, tid);

  for (int n = 0; n < NT; ++n) {
    const int t0 = n * BT;
    const int cur = n & 1;
    float* Sq = SqB + cur * (BT * QS);
    float* Sk = SkB + cur * (BT * QS);
    float* Sv = SvB + cur * (BT * VST);

    if (n + 1 < NT) {
      const int nb = (n + 1) & 1;
      issue_chunk_loads(qb, kb, vp, t0 + BT,
                        SqB + nb * (BT * QS), SkB + nb * (BT * QS),
                        SvB + nb * (BT * VST), tid);
    }

    // ---- gate precompute: wave0 shfl-scan of g over the chunk ----
    if (wave == 0) {
      float x0 = gb[t0 + 2 * lane];
      float x1 = gb[t0 + 2 * lane + 1];
      float p = x0 + x1;
#pragma unroll
      for (int off = 1; off < 32; off <<= 1) {
        float y = __shfl_up(p, off, 32);
        if (lane >= off) p += y;
      }
      float c1 = p;        // gc[2l+1]
      float c0 = p - x1;   // gc[2l]
      float glast = __shfl(p, 31, 32);
      EgS[2 * lane]     = __expf(c0) * scale;
      EgS[2 * lane + 1] = __expf(c1) * scale;
      Rg[2 * lane]      = __expf(-c0);
      Rg[2 * lane + 1]  = __expf(-c1);
      Wv[2 * lane]      = __expf(glast - c0);
      Wv[2 * lane + 1]  = __expf(glast - c1);
      if (lane == 31) Dec = __expf(glast);
    }

    // wait for chunk n's async data (chunk n+1's 34 ops may stay in flight)
    if (n + 1 < NT) asm volatile("s_wait_asynccnt 0x22" ::: "memory");
    else            asm volatile("s_wait_asynccnt 0x0"  ::: "memory");
    __syncthreads();

    // ---- o_inter: q @ S_entering, scaled afterwards by e^{gc_i}*scale ----
    v8f acc = {};
#pragma unroll 4
    for (int k0 = 0; k0 < KD; k0 += 4) {
      v2f a = ld_a(Sq, QS, m0, k0, lr, hf);
      v2f b = ld_b(St, VST, k0, 0, lr, hf);
      acc = wmma4(a, b, acc);
    }
#pragma unroll
    for (int r = 0; r < 8; ++r) acc[r] *= EgS[m0 + r + 8 * hf];

    // ---- s = q@k^T over causal j-tiles, reusing each q A-frag ----
    v8f sj[4] = {};
#pragma unroll 4
    for (int k0 = 0; k0 < KD; k0 += 4) {
      v2f a = ld_a(Sq, QS, m0, k0, lr, hf);
      for (int jt = 0; jt <= wave; ++jt) {
        v2f b = ld_a(Sk, QS, jt * 16, k0, lr, hf);  // k^T B-frag == k A-frag
        sj[jt] = wmma4(a, b, sj[jt]);
      }
    }

    // ---- gate+mask each s tile, o += s_tile @ v ----
    for (int jt = 0; jt <= wave; ++jt) {
      const int j0 = jt * 16;
      const float rj = Rg[j0 + lr];
#pragma unroll
      for (int r = 0; r < 8; ++r) {
        int i = m0 + r + 8 * hf;
        float val = (i >= j0 + lr) ? sj[jt][r] * EgS[i] * rj : 0.0f;
        slab[(r + 8 * hf) * SSL + lr] = val;
      }
#pragma unroll
      for (int jc = 0; jc < 16; jc += 4) {
        v2f a = ld_a(slab, SSL, 0, jc, lr, hf);
        v2f b;
        b.x = Sv[(j0 + jc + 2 * hf) * VST + lr];
        b.y = Sv[(j0 + jc + 2 * hf + 1) * VST + lr];
        acc = wmma4(a, b, acc);
      }
    }

    // ---- store output tile ----
#pragma unroll
    for (int r = 0; r < 8; ++r) {
      int i = m0 + r + 8 * hf;
      ob[(size_t)(t0 + i) * VD + lr] = acc[r];
    }

    __syncthreads();  // o_inter reads of St complete before update

    // ---- state update: S = Dec*S + k^T @ (v*Wv); wave owns k-tiles 2w,2w+1 ----
#pragma unroll
    for (int kt = 0; kt < 2; ++kt) {
      const int kr0 = (2 * wave + kt) * 16;
      v8f kv = {};
#pragma unroll 4
      for (int j0 = 0; j0 < BT; j0 += 4) {
        v2f a;  // A = k^T: A[i][jj] = k[j0+jj][kr0+i]
        a.x = Sk[(j0 + 2 * hf) * QS + kr0 + lr];
        a.y = Sk[(j0 + 2 * hf + 1) * QS + kr0 + lr];
        v2f b;  // B = (v * Wv)[j][col]
        b.x = Sv[(j0 + 2 * hf) * VST + lr] * Wv[j0 + 2 * hf];
        b.y = Sv[(j0 + 2 * hf + 1) * VST + lr] * Wv[j0 + 2 * hf + 1];
        kv = wmma4(a, b, kv);
      }
      const float dec = Dec;
#pragma unroll
      for (int r = 0; r < 8; ++r) {
        int row = kr0 + r + 8 * hf;
        St[row * VST + lr] = dec * St[row * VST + lr] + kv[r];
      }
    }
    __syncthreads();  // state ready, buffers free before next chunk
  }
}

extern "C" void kernel_launch(void* const* d_in, const int* in_sizes, int n_in,
                              void* d_out, int out_size, void* d_ws, size_t ws_size,
                              hipStream_t stream) {
  const float* q = (const float*)d_in[0];
  const float* k = (const float*)d_in[1];
  const float* v = (const float*)d_in[2];
  const float* g = (const float*)d_in[3];
  float* out = (float*)d_out;

  const int BH = in_sizes[3] / T_LEN;  // B*H from g's element count
  dim3 grid(BH * NVB);
  dim3 block(128);
  gla_fwd_kernel<<<grid, block, 0, stream>>>(q, k, v, g, out);
}